// GeometricConstraintMessagePassing_27023934227041
// MI455X (gfx1250) — compile-verified
//
#include <hip/hip_runtime.h>
#include <hip/hip_bf16.h>

#define N_NODES 2048
#define N_EDGES 32768
#define HDIM    128
#define KMSG    260           // 2*H + 4 edge features
#define XSTR    268           // LDS row stride for 260-wide tile (bank-conflict friendly, 16B aligned)
#define HSTR    132           // LDS row stride for 128/129-wide tiles

typedef __attribute__((ext_vector_type(2))) float v2f;
typedef __attribute__((ext_vector_type(8))) float v8f;

__device__ __forceinline__ v8f wmma4_f32(v2f a, v2f b, v8f c) {
    // D = A(16x4 f32) * B(4x16 f32) + C(16x16 f32)
    return __builtin_amdgcn_wmma_f32_16x16x4_f32(false, a, false, b, (short)0, c, false, false);
}

// ---------------------------------------------------------------------------
// Step 2: scatter edge distances into dense adjacency (adj pre-filled < 0)
// ---------------------------------------------------------------------------
__global__ void scatter_adj_kernel(const int* __restrict__ eidx,
                                   const float* __restrict__ edist,
                                   float* __restrict__ adj) {
    int e = blockIdx.x * blockDim.x + threadIdx.x;
    if (e < N_EDGES) {
        int s = eidx[e];
        int d = eidx[N_EDGES + e];
        adj[s * N_NODES + d] = edist[e];
    }
}

// ---------------------------------------------------------------------------
// Step 3: two-hop residuals. One wave32 per edge scans all N intermediates.
// ---------------------------------------------------------------------------
__global__ void residual_kernel(const int* __restrict__ eidx,
                                const float* __restrict__ edist,
                                const float* __restrict__ adj,
                                float* __restrict__ r_out,
                                float* __restrict__ wts,
                                float* __restrict__ rsum,
                                float* __restrict__ wsum,
                                float* __restrict__ cnt) {
    int wave = threadIdx.x >> 5;
    int lane = threadIdx.x & 31;
    int e = blockIdx.x * 8 + wave;
    int s = eidx[e];
    int d = eidx[N_EDGES + e];
    float dac = edist[e];

    const float* rowA = adj + (size_t)s * N_NODES;
    float sum = 0.f, count = 0.f;
    for (int b = lane; b < N_NODES; b += 32) {
        float dab = rowA[b];
        float dbc = adj[(size_t)b * N_NODES + d];
        if (dab >= 0.f && dbc >= 0.f) { sum += dab + dbc; count += 1.f; }
    }
    // wave32 reduction
    for (int off = 16; off > 0; off >>= 1) {
        sum   += __shfl_down(sum, off, 32);
        count += __shfl_down(count, off, 32);
    }
    if (lane == 0) {
        float mean = (count > 0.f) ? (sum / fmaxf(count, 1.f)) : dac;
        float r = fabsf(dac - mean);
        float w = __expf(-r);
        r_out[e] = r;
        wts[e] = w;
        atomicAdd(&rsum[d], r);
        atomicAdd(&wsum[d], w);
        atomicAdd(&cnt[d], 1.f);
    }
}

// ---------------------------------------------------------------------------
// Step 4: edge MLP via f32 WMMA. One wave per 32 edges (M=32 tile):
// two 16-row accumulators share each B-fragment load -> 2 WMMAs per B pair.
//   X(32x260) -> relu(X@W1+b1)(32x128) -> (@W2+b2)*w -> atomic scatter to agg
// ---------------------------------------------------------------------------
__global__ void __launch_bounds__(32)
edge_mlp_kernel(const float* __restrict__ mu, const float* __restrict__ sigma,
                const int* __restrict__ eidx,
                const float* __restrict__ ed, const float* __restrict__ ec,
                const float* __restrict__ ea, const float* __restrict__ edd,
                const float* __restrict__ w1, const float* __restrict__ b1,
                const float* __restrict__ w2, const float* __restrict__ b2,
                const float* __restrict__ wts,
                float* __restrict__ agg) {
    __shared__ float X[32][XSTR];
    __shared__ float Hd[32][HSTR];

    const int lane = threadIdx.x;
    const int e0 = blockIdx.x * 32;
    const int* src = eidx;
    const int* dst = eidx + N_EDGES;

    // Gather mu[src] | sigma[src] | edge feats into LDS (float4 per lane)
    for (int r = 0; r < 32; ++r) {
        int s = src[e0 + r];
        ((float4*)&X[r][0])[lane]    = ((const float4*)(mu    + (size_t)s * HDIM))[lane];
        ((float4*)&X[r][HDIM])[lane] = ((const float4*)(sigma + (size_t)s * HDIM))[lane];
    }
    {
        int e = e0 + lane;
        X[lane][256] = ed[e];
        X[lane][257] = ec[e];
        X[lane][258] = ea[e];
        X[lane][259] = edd[e];
    }
    __syncthreads();

    const int m   = lane & 15;
    const int hi  = lane >> 4;      // 0: K {0,1}, 1: K {2,3}
    const int kA0 = hi * 2;

    // ---- layer 1: hidden = relu(X @ W1 + b1), K = 260 ----
    for (int nt = 0; nt < 8; ++nt) {
        const int n = nt * 16 + m;
        v8f acc0 = {};   // edge rows 0..15
        v8f acc1 = {};   // edge rows 16..31
        for (int k = 0; k < KMSG; k += 4) {
            v2f a0, a1, b;
            b.x  = w1[(size_t)(k + kA0)     * HDIM + n];
            b.y  = w1[(size_t)(k + kA0 + 1) * HDIM + n];
            a0.x = X[m][k + kA0];
            a0.y = X[m][k + kA0 + 1];
            a1.x = X[16 + m][k + kA0];
            a1.y = X[16 + m][k + kA0 + 1];
            acc0 = wmma4_f32(a0, b, acc0);
            acc1 = wmma4_f32(a1, b, acc1);
        }
        float bias = b1[n];
        for (int v = 0; v < 8; ++v) {
            int row = v + hi * 8;
            Hd[row][n]      = fmaxf(acc0[v] + bias, 0.f);
            Hd[16 + row][n] = fmaxf(acc1[v] + bias, 0.f);
        }
    }
    __syncthreads();

    // ---- layer 2: msg = hidden @ W2 + b2; scatter w*msg to agg[dst] ----
    for (int nt = 0; nt < 8; ++nt) {
        const int n = nt * 16 + m;
        v8f acc0 = {};
        v8f acc1 = {};
        for (int k = 0; k < HDIM; k += 4) {
            v2f a0, a1, b;
            b.x  = w2[(size_t)(k + kA0)     * HDIM + n];
            b.y  = w2[(size_t)(k + kA0 + 1) * HDIM + n];
            a0.x = Hd[m][k + kA0];
            a0.y = Hd[m][k + kA0 + 1];
            a1.x = Hd[16 + m][k + kA0];
            a1.y = Hd[16 + m][k + kA0 + 1];
            acc0 = wmma4_f32(a0, b, acc0);
            acc1 = wmma4_f32(a1, b, acc1);
        }
        float bias = b2[n];
        for (int v = 0; v < 8; ++v) {
            int row = v + hi * 8;
            int eA = e0 + row;
            int eB = e0 + 16 + row;
            float valA = (acc0[v] + bias) * wts[eA];
            float valB = (acc1[v] + bias) * wts[eB];
            atomicAdd(&agg[(size_t)dst[eA] * HDIM + n], valA);
            atomicAdd(&agg[(size_t)dst[eB] * HDIM + n], valB);
        }
    }
}

// ---------------------------------------------------------------------------
// Step 5: node updates (mu MLP + sigma MLP), one wave per 16 nodes.
// ---------------------------------------------------------------------------
__global__ void __launch_bounds__(32)
node_mlp_kernel(const float* __restrict__ mu,
                const float* __restrict__ agg_in,
                const float* __restrict__ wsum,
                const float* __restrict__ rsum,
                const float* __restrict__ cnt,
                const float* __restrict__ uw1, const float* __restrict__ ub1,
                const float* __restrict__ uw2, const float* __restrict__ ub2,
                const float* __restrict__ sw1, const float* __restrict__ sb1,
                const float* __restrict__ sw2, const float* __restrict__ sb2,
                float* __restrict__ mu_out, float* __restrict__ sig_out) {
    __shared__ float Xa[16][HSTR];   // normalized agg (cols 0..127), col 128 = mean_r, 129..131 = 0
    __shared__ float Hd[16][HSTR];

    const int lane = threadIdx.x;
    const int n0 = blockIdx.x * 16;

    for (int r = 0; r < 16; ++r) {
        int node = n0 + r;
        float ws = fmaxf(wsum[node], 1e-8f);
        float4 v = ((const float4*)(agg_in + (size_t)node * HDIM))[lane];
        v.x /= ws; v.y /= ws; v.z /= ws; v.w /= ws;
        ((float4*)&Xa[r][0])[lane] = v;
    }
    if (lane < 16) {
        int node = n0 + lane;
        Xa[lane][128] = rsum[node] / fmaxf(cnt[node], 1.f);
        Xa[lane][129] = 0.f; Xa[lane][130] = 0.f; Xa[lane][131] = 0.f;
    }
    __syncthreads();

    const int m   = lane & 15;
    const int hi  = lane >> 4;
    const int kA0 = hi * 2;

    // ===================== mu MLP =====================
    for (int nt = 0; nt < 8; ++nt) {
        const int n = nt * 16 + m;
        v8f acc = {};
        for (int k = 0; k < HDIM; k += 4) {
            v2f a, b;
            a.x = Xa[m][k + kA0];
            a.y = Xa[m][k + kA0 + 1];
            b.x = uw1[(size_t)(k + kA0)     * HDIM + n];
            b.y = uw1[(size_t)(k + kA0 + 1) * HDIM + n];
            acc = wmma4_f32(a, b, acc);
        }
        float bias = ub1[n];
        for (int v = 0; v < 8; ++v) Hd[v + hi * 8][n] = fmaxf(acc[v] + bias, 0.f);
    }
    __syncthreads();
    for (int nt = 0; nt < 8; ++nt) {
        const int n = nt * 16 + m;
        v8f acc = {};
        for (int k = 0; k < HDIM; k += 4) {
            v2f a, b;
            a.x = Hd[m][k + kA0];
            a.y = Hd[m][k + kA0 + 1];
            b.x = uw2[(size_t)(k + kA0)     * HDIM + n];
            b.y = uw2[(size_t)(k + kA0 + 1) * HDIM + n];
            acc = wmma4_f32(a, b, acc);
        }
        float bias = ub2[n];
        for (int v = 0; v < 8; ++v) {
            int row = v + hi * 8;
            size_t o = (size_t)(n0 + row) * HDIM + n;
            mu_out[o] = mu[o] + acc[v] + bias;
        }
    }
    __syncthreads();

    // ===================== sigma MLP (K = 129) =====================
    for (int nt = 0; nt < 8; ++nt) {
        const int n = nt * 16 + m;
        v8f acc = {};
        for (int k = 0; k < HDIM; k += 4) {
            v2f a, b;
            a.x = Xa[m][k + kA0];
            a.y = Xa[m][k + kA0 + 1];
            b.x = sw1[(size_t)(k + kA0)     * HDIM + n];
            b.y = sw1[(size_t)(k + kA0 + 1) * HDIM + n];
            acc = wmma4_f32(a, b, acc);
        }
        {   // K tail: row 128 (mean_r feature); rows 129..131 zero-padded
            v2f a, b;
            a.x = Xa[m][128 + kA0];          // zero for hi==1
            a.y = Xa[m][129 + kA0];          // always zero
            b.x = (hi == 0) ? sw1[(size_t)128 * HDIM + n] : 0.f;
            b.y = 0.f;
            acc = wmma4_f32(a, b, acc);
        }
        float bias = sb1[n];
        for (int v = 0; v < 8; ++v) Hd[v + hi * 8][n] = fmaxf(acc[v] + bias, 0.f);
    }
    __syncthreads();
    for (int nt = 0; nt < 8; ++nt) {
        const int n = nt * 16 + m;
        v8f acc = {};
        for (int k = 0; k < HDIM; k += 4) {
            v2f a, b;
            a.x = Hd[m][k + kA0];
            a.y = Hd[m][k + kA0 + 1];
            b.x = sw2[(size_t)(k + kA0)     * HDIM + n];
            b.y = sw2[(size_t)(k + kA0 + 1) * HDIM + n];
            acc = wmma4_f32(a, b, acc);
        }
        float bias = sb2[n];
        for (int v = 0; v < 8; ++v) {
            int row = v + hi * 8;
            float x = acc[v] + bias;
            float sp = (x > 20.f) ? x : log1pf(__expf(x));   // softplus
            sig_out[(size_t)(n0 + row) * HDIM + n] = sp;
        }
    }
}

// ---------------------------------------------------------------------------
extern "C" void kernel_launch(void* const* d_in, const int* in_sizes, int n_in,
                              void* d_out, int out_size, void* d_ws, size_t ws_size,
                              hipStream_t stream) {
    const float* mu    = (const float*)d_in[0];
    const float* sigma = (const float*)d_in[1];
    const int*   eidx  = (const int*)  d_in[2];
    const float* ed    = (const float*)d_in[3];
    const float* ec    = (const float*)d_in[4];
    const float* ea    = (const float*)d_in[5];
    const float* edd   = (const float*)d_in[6];
    const float* mw1   = (const float*)d_in[7];
    const float* mb1   = (const float*)d_in[8];
    const float* mw2   = (const float*)d_in[9];
    const float* mb2   = (const float*)d_in[10];
    const float* uw1   = (const float*)d_in[11];
    const float* ub1   = (const float*)d_in[12];
    const float* uw2   = (const float*)d_in[13];
    const float* ub2   = (const float*)d_in[14];
    const float* sw1   = (const float*)d_in[15];
    const float* sb1   = (const float*)d_in[16];
    const float* sw2   = (const float*)d_in[17];
    const float* sb2   = (const float*)d_in[18];

    // Workspace layout
    char*  ws   = (char*)d_ws;
    float* adj  = (float*)ws;                                          // N*N
    float* agg  = (float*)(ws + (size_t)N_NODES * N_NODES * 4);        // N*H
    float* wsum = agg + (size_t)N_NODES * HDIM;                        // N
    float* rsum = wsum + N_NODES;                                      // N
    float* cnt  = rsum + N_NODES;                                      // N
    float* wts  = cnt + N_NODES;                                       // E

    float* mu_out  = (float*)d_out;
    float* sig_out = mu_out + (size_t)N_NODES * HDIM;
    float* r_out   = sig_out + (size_t)N_NODES * HDIM;

    // adj <- 0xBFBFBFBF per float (~ -1.498): any negative marker works,
    // validity test in reference is (d >= 0). Accumulators <- 0.
    hipMemsetAsync(adj, 0xBF, (size_t)N_NODES * N_NODES * 4, stream);
    hipMemsetAsync(agg, 0, (size_t)(N_NODES * HDIM + 3 * N_NODES) * 4, stream);

    scatter_adj_kernel<<<N_EDGES / 256, 256, 0, stream>>>(eidx, ed, adj);
    residual_kernel<<<N_EDGES / 8, 256, 0, stream>>>(eidx, ed, adj, r_out, wts, rsum, wsum, cnt);
    edge_mlp_kernel<<<N_EDGES / 32, 32, 0, stream>>>(mu, sigma, eidx, ed, ec, ea, edd,
                                                     mw1, mb1, mw2, mb2, wts, agg);
    node_mlp_kernel<<<N_NODES / 16, 32, 0, stream>>>(mu, agg, wsum, rsum, cnt,
                                                     uw1, ub1, uw2, ub2,
                                                     sw1, sb1, sw2, sb2,
                                                     mu_out, sig_out);
}